// GNN_76905684402542
// MI455X (gfx1250) — compile-verified
//
#include <hip/hip_runtime.h>

typedef float v2f __attribute__((ext_vector_type(2)));
typedef float v8f __attribute__((ext_vector_type(8)));

#define IN_DIM  64
#define HID_DIM 64
#define OUT_DIM 32

// ---------------- degree / symmetric normalization ----------------
__global__ void k_deg_init(float* __restrict__ deg, int N) {
    int i = blockIdx.x * blockDim.x + threadIdx.x;
    if (i < N) deg[i] = 1.0f;   // self-loop contributes 1
}

__global__ void k_deg_count(const int* __restrict__ dst, float* __restrict__ deg, int E) {
    int e = blockIdx.x * blockDim.x + threadIdx.x;
    if (e < E) atomicAdd(&deg[dst[e]], 1.0f);
}

__global__ void k_dinv(float* __restrict__ deg, int N) {
    int i = blockIdx.x * blockDim.x + threadIdx.x;
    if (i < N) deg[i] = rsqrtf(deg[i]);   // deg >= 1 always (self-loops)
}

// ---------------- dense GEMM via V_WMMA_F32_16X16X4_F32 ----------------
// One wave computes a 16-row x NC-col tile of  out = (RELU? relu(A) : A) @ W.
// A: [N,64] row-major, W: [64,NC] row-major. K is always 64 here.
// fp32 WMMA A-frag layout (ISA 7.12.2): lanes 0-15 hold K={0,1}, lanes 16-31 hold K={2,3};
// C/D: 8 VGPRs, lane%16 = column, vgpr v = row v (lanes<16) or v+8 (lanes>=16).
template <int NC, bool RELU>
__global__ void k_gemm_wmma(const float* __restrict__ A, const float* __restrict__ W,
                            float* __restrict__ out, int nTiles) {
    const int lane = threadIdx.x & 31;
    const int wave = threadIdx.x >> 5;
    const int tile = blockIdx.x * (blockDim.x >> 5) + wave;
    if (tile >= nTiles) return;                 // uniform per wave: EXEC stays all-1s

    const int half = lane >> 4;                 // 0: K pair {0,1}, 1: K pair {2,3}
    const int l16  = lane & 15;
    const float* __restrict__ arow = A + (size_t)(tile * 16 + l16) * 64;

    constexpr int NT = NC / 16;
    v8f acc[NT] = {};

#pragma unroll
    for (int k = 0; k < 64; k += 4) {
        const int kk = k + 2 * half;
        float a0 = arow[kk];
        float a1 = arow[kk + 1];
        if (RELU) { a0 = fmaxf(a0, 0.0f); a1 = fmaxf(a1, 0.0f); }
        v2f a; a.x = a0; a.y = a1;
#pragma unroll
        for (int j = 0; j < NT; ++j) {
            v2f b;
            b.x = W[kk * NC + j * 16 + l16];
            b.y = W[(kk + 1) * NC + j * 16 + l16];
            acc[j] = __builtin_amdgcn_wmma_f32_16x16x4_f32(
                false, a, false, b, (short)0, acc[j], false, false);
        }
    }

#pragma unroll
    for (int j = 0; j < NT; ++j) {
#pragma unroll
        for (int v = 0; v < 8; ++v) {
            const int r = tile * 16 + v + 8 * half;
            out[(size_t)r * NC + j * 16 + l16] = acc[j][v];
        }
    }
}

// scalar tail for N % 16 rows (not hit for N=100000, kept for safety)
template <int NC, bool RELU>
__global__ void k_gemm_tail(const float* __restrict__ A, const float* __restrict__ W,
                            float* __restrict__ out, int rowStart, int N) {
    int gid = blockIdx.x * blockDim.x + threadIdx.x;
    int r = rowStart + gid / NC;
    int c = gid % NC;
    if (r >= N) return;
    float s = 0.0f;
    for (int k = 0; k < 64; ++k) {
        float a = A[(size_t)r * 64 + k];
        if (RELU) a = fmaxf(a, 0.0f);
        s += a * W[k * NC + c];
    }
    out[(size_t)r * NC + c] = s;
}

// ---------------- aggregation: self-loop + bias init, then edge scatter ----------------
template <int C>
__global__ void k_init_agg(const float* __restrict__ h, const float* __restrict__ dinv,
                           const float* __restrict__ bias, float* __restrict__ agg, int N) {
    constexpr int SH = (C == 64) ? 6 : 5;
    long long gid = (long long)blockIdx.x * blockDim.x + threadIdx.x;
    if (gid >= (long long)N * C) return;
    int i = (int)(gid >> SH);
    int c = (int)(gid & (C - 1));
    float di = dinv[i];
    agg[gid] = di * di * h[gid] + bias[c];      // self-loop norm = dinv^2, plus bias
}

template <int C>
__global__ void k_scatter(const float* __restrict__ h, const int* __restrict__ src,
                          const int* __restrict__ dst, const float* __restrict__ dinv,
                          float* __restrict__ agg, int E) {
    constexpr int SH = (C == 64) ? 6 : 5;
    long long gid = (long long)blockIdx.x * blockDim.x + threadIdx.x;
    int e = (int)(gid >> SH);
    if (e >= E) return;
    int c = (int)(gid & (C - 1));
    int s = src[e];
    int d = dst[e];
    float w = dinv[s] * dinv[d];
    atomicAdd(&agg[(size_t)d * C + c], w * h[(size_t)s * C + c]);
}

// ---------------- driver ----------------
extern "C" void kernel_launch(void* const* d_in, const int* in_sizes, int n_in,
                              void* d_out, int out_size, void* d_ws, size_t ws_size,
                              hipStream_t stream) {
    const float* x  = (const float*)d_in[0];
    const int*   ei = (const int*)d_in[1];
    const float* W1 = (const float*)d_in[2];
    const float* b1 = (const float*)d_in[3];
    const float* W2 = (const float*)d_in[4];
    const float* b2 = (const float*)d_in[5];

    const int N = in_sizes[0] / IN_DIM;
    const int E = in_sizes[1] / 2;
    const int* src = ei;        // edge_index[0]
    const int* dst = ei + E;    // edge_index[1]

    float* ws   = (float*)d_ws;
    float* dinv = ws;                                   // [N] degrees -> dinv
    float* buf1 = ws + (((size_t)N + 63) & ~(size_t)63); // [N*64] h0, then reused for h2 [N*32]
    float* buf2 = buf1 + (size_t)N * HID_DIM;            // [N*64] aggregated layer-1 (pre-relu)
    float* out  = (float*)d_out;                         // [N*32]

    const int B = 256;

    // degrees + normalization
    k_deg_init<<<(N + B - 1) / B, B, 0, stream>>>(dinv, N);
    k_deg_count<<<(E + B - 1) / B, B, 0, stream>>>(dst, dinv, E);
    k_dinv<<<(N + B - 1) / B, B, 0, stream>>>(dinv, N);

    const int nTiles = N / 16;
    const int rem    = N % 16;

    // ---- layer 1: h0 = x @ W1 ; agg1 = scatter(norm * h0) + b1 ----
    k_gemm_wmma<HID_DIM, false><<<(nTiles + 3) / 4, 128, 0, stream>>>(x, W1, buf1, nTiles);
    if (rem)
        k_gemm_tail<HID_DIM, false><<<(rem * HID_DIM + B - 1) / B, B, 0, stream>>>(
            x, W1, buf1, nTiles * 16, N);
    {
        long long tot = (long long)N * HID_DIM;
        k_init_agg<HID_DIM><<<(int)((tot + B - 1) / B), B, 0, stream>>>(buf1, dinv, b1, buf2, N);
        long long stot = (long long)E * HID_DIM;
        k_scatter<HID_DIM><<<(int)((stot + B - 1) / B), B, 0, stream>>>(buf1, src, dst, dinv, buf2, E);
    }

    // ---- layer 2: h2 = relu(agg1) @ W2 (relu folded into A loads) ----
    k_gemm_wmma<OUT_DIM, true><<<(nTiles + 3) / 4, 128, 0, stream>>>(buf2, W2, buf1, nTiles);
    if (rem)
        k_gemm_tail<OUT_DIM, true><<<(rem * OUT_DIM + B - 1) / B, B, 0, stream>>>(
            buf2, W2, buf1, nTiles * 16, N);
    {
        long long tot = (long long)N * OUT_DIM;
        k_init_agg<OUT_DIM><<<(int)((tot + B - 1) / B), B, 0, stream>>>(buf1, dinv, b2, out, N);
        long long stot = (long long)E * OUT_DIM;
        k_scatter<OUT_DIM><<<(int)((stot + B - 1) / B), B, 0, stream>>>(buf1, src, dst, dinv, out, E);
    }
}